// VotingLayer_74010876444938
// MI455X (gfx1250) — compile-verified
//
#include <hip/hip_runtime.h>
#include <stdint.h>

// ---------------------------------------------------------------------------
// Sinkhorn normalization of 131072 independent 16x16 matrices (10 iterations),
// reformulated as diagonal-scaling recurrence:
//     u0 = 1;  repeat 10x: { w = 1/(S^T u);  u = 1/(S w); }
//     out[i][j] = S[i][j] * u[i] * w[j]
// which is algebraically identical to the reference's alternating
// column/row renormalization (scales cancel telescopically).
//
// One matrix per wave32. Matvecs run on the matrix pipe as chained
// V_WMMA_F32_16X16X4_F32 with replicated-vector operands:
//   T = U_rep x S   -> every row of T equals t = S^T u   (D-layout)
//   E = S x W_rep   -> every col of E equals s = S w     (D-layout)
// VALU does only rcp (+1 Newton step) and operand replication via
// v_readlane / cndmask; WMMA co-executes with VALU (ISA 8.4: XDL ops
// tracked as TRANS).
//
// Data movement: per-wave GLOBAL_LOAD_ASYNC_TO_LDS_B128 (2 x 512B, fully
// coalesced), s_wait_asynccnt 0 (wave-local staging -> no block barrier),
// then ds_read gathers build S in both WMMA A-layout (16x4 column chunks)
// and B-layout (4x16 row chunks) register images.
// ---------------------------------------------------------------------------

typedef float v8f __attribute__((ext_vector_type(8)));
typedef float v2f __attribute__((ext_vector_type(2)));

__device__ __forceinline__ float lane_bcast(float v, int srclane) {
  return __builtin_bit_cast(
      float, __builtin_amdgcn_readlane(__builtin_bit_cast(int, v), srclane));
}

__device__ __forceinline__ float fast_rcp(float x) {
  float r = __builtin_amdgcn_rcpf(x);
  r = r * (2.0f - x * r);   // one Newton step -> ~0.5 ulp vs exact divide
  return r;
}

__global__ __launch_bounds__(256)
void sinkhorn16_wmma(const float* __restrict__ v, float* __restrict__ out,
                     int batch) {
  __shared__ float tile[8 * 256];           // 1 KB per wave, 8 waves/block

  const int lane = threadIdx.x & 31;
  const int wave = threadIdx.x >> 5;
  const int m    = blockIdx.x * 8 + wave;   // matrix handled by this wave
  const int mc   = (m < batch) ? m : (batch - 1);  // clamp tail loads
  const int h    = lane >> 4;               // 0 = lanes 0-15, 1 = lanes 16-31
  const int l15  = lane & 15;
  const bool hi  = (h != 0);

  // ---- async global -> LDS: 1 KB matrix, wave-local (no block barrier) ----
  // NOTE: INST_OFFSET is applied to BOTH the LDS and global address
  // (CDNA5 ISA), so one voffset/ldsaddr pair serves both halves.
  uint32_t lds_byte =
      (uint32_t)(uintptr_t)(&tile[wave * 256]) + (uint32_t)(lane * 16);
  int gvoff = mc * 1024 + lane * 16;        // < 2^31: 131072 * 1KB = 134 MB
  asm volatile(
      "global_load_async_to_lds_b128 %0, %1, %2 offset:0\n\t"
      "global_load_async_to_lds_b128 %0, %1, %2 offset:512\n\t"
      "s_wait_asynccnt 0"
      :
      : "v"(lds_byte), "v"(gvoff), "s"(v)
      : "memory");

  // LDS image: s[col*16 + row] = S[row][col]  (v is column-major in S)
  const float* s = &tile[wave * 256];

  // ---- build S in WMMA A-layout (column chunks) and B-layout (row chunks) --
  // A 16x4 f32 layout: v0 = {lo: K+0, hi: K+2}, v1 = {lo: K+1, hi: K+3},
  //                    lane(mod 16) = M.
  // B  4x16 f32 layout: v0 = {lo: row K+0, hi: row K+1},
  //                     v1 = {lo: row K+2, hi: row K+3}, lane(mod 16) = N.
  v2f As[4], Bs[4];
#pragma unroll
  for (int c = 0; c < 4; ++c) {
    As[c].x = s[(4 * c + 0 + 2 * h) * 16 + l15] + 1e-6f;
    As[c].y = s[(4 * c + 1 + 2 * h) * 16 + l15] + 1e-6f;
    Bs[c].x = s[l15 * 16 + (4 * c + 0 + h)] + 1e-6f;
    Bs[c].y = s[l15 * 16 + (4 * c + 2 + h)] + 1e-6f;
  }

  // ---- Sinkhorn scale recurrence ------------------------------------------
  v2f Au[4];                                 // u replicated in A-layout
#pragma unroll
  for (int c = 0; c < 4; ++c) { Au[c].x = 1.0f; Au[c].y = 1.0f; }

  float w = 1.0f;                            // per-lane w[lane & 15]
  float su[16];                              // wave-uniform u[i] (SGPRs)

#pragma unroll
  for (int it = 0; it < 10; ++it) {
    // T = U_rep x S : every row of T equals t[j] = sum_i u[i] S[i][j]
    v8f T = {};
#pragma unroll
    for (int c = 0; c < 4; ++c)
      T = __builtin_amdgcn_wmma_f32_16x16x4_f32(false, Au[c], false, Bs[c],
                                                (short)0, T, false, false);
    // w[j] = 1/t[j]; T reg0 holds t[lane&15] in every lane (both halves)
    float t0 = T[0];
    w = fast_rcp(t0);

    // replicate w into B-layout operand via readlane (wave-uniform SGPRs)
    float sw[16];
#pragma unroll
    for (int j = 0; j < 16; ++j) sw[j] = lane_bcast(w, j);
    v2f Bw[4];
#pragma unroll
    for (int c = 0; c < 4; ++c) {
      Bw[c].x = hi ? sw[4 * c + 1] : sw[4 * c + 0];
      Bw[c].y = hi ? sw[4 * c + 3] : sw[4 * c + 2];
    }

    // E = S x W_rep : every column of E equals s[i] = sum_j S[i][j] w[j]
    v8f E = {};
#pragma unroll
    for (int c = 0; c < 4; ++c)
      E = __builtin_amdgcn_wmma_f32_16x16x4_f32(false, As[c], false, Bw[c],
                                                (short)0, E, false, false);

    // u[i] = 1/s[i]; E reg r holds s[r] (lo lanes) / s[r+8] (hi lanes)
    float ur[8];
#pragma unroll
    for (int r = 0; r < 8; ++r) ur[r] = fast_rcp(E[r]);
#pragma unroll
    for (int r = 0; r < 8; ++r) {
      su[r]     = lane_bcast(ur[r], 0);
      su[r + 8] = lane_bcast(ur[r], 16);
    }
    // replicate u into A-layout operand
#pragma unroll
    for (int c = 0; c < 4; ++c) {
      Au[c].x = hi ? su[4 * c + 2] : su[4 * c + 0];
      Au[c].y = hi ? su[4 * c + 3] : su[4 * c + 1];
    }
  }

  // ---- output: out[i][j] = S[i][j] * u[i] * w[j], coalesced b32 stores ----
  if (m < batch) {
    float* o = out + m * 256;
#pragma unroll
    for (int c = 0; c < 4; ++c) {
      // Bs[c].x holds row (4c+h), Bs[c].y holds row (4c+2+h); col = l15
      float u0 = hi ? su[4 * c + 1] : su[4 * c + 0];
      float u1 = hi ? su[4 * c + 3] : su[4 * c + 2];
      o[(4 * c + 0 + h) * 16 + l15] = Bs[c].x * u0 * w;
      o[(4 * c + 2 + h) * 16 + l15] = Bs[c].y * u1 * w;
    }
  }
}

extern "C" void kernel_launch(void* const* d_in, const int* in_sizes, int n_in,
                              void* d_out, int out_size, void* d_ws,
                              size_t ws_size, hipStream_t stream) {
  const float* v = (const float*)d_in[0];
  float* out = (float*)d_out;
  const int batch = in_sizes[0] / 256;          // 131072 matrices
  const int blocks = (batch + 7) / 8;           // 8 waves (matrices) per block
  sinkhorn16_wmma<<<blocks, 256, 0, stream>>>(v, out, batch);
}